// GCN_64845416235488
// MI455X (gfx1250) — compile-verified
//
#include <hip/hip_runtime.h>
#include <hip/hip_bf16.h>

typedef __attribute__((ext_vector_type(2))) float v2f;
typedef __attribute__((ext_vector_type(8))) float v8f;

#define NODES 50000
#define EDGES 800000

// ---------------------------------------------------------------------------
// GEMM: C[M,N] = A[M,K] @ B[K,N], fp32, one wave per 16x16 output tile using
// V_WMMA_F32_16X16X4_F32 (K stepped by 4).  M multiple of 16 (50000=3125*16).
// FULLN=true: N multiple of 16, no column guards (branch-free inner loop).
// FULLN=false: column clamp + 0/1 scale, still branch-free (no EXEC churn).
// ---------------------------------------------------------------------------
template <bool FULLN>
__global__ __launch_bounds__(256) void gemm_f32_wmma(
    const float* __restrict__ A, const float* __restrict__ B,
    float* __restrict__ C, int M, int N, int K, int tilesN)
{
    const int wave = threadIdx.x >> 5;
    const int lane = threadIdx.x & 31;
    const int tile = blockIdx.x * 8 + wave;
    const int tilesM = M >> 4;
    if (tile >= tilesM * tilesN) return;      // wave-uniform: EXEC stays all-1s

    const int tm   = tile / tilesN;
    const int tn   = tile % tilesN;
    const int half = lane >> 4;               // 0: lanes 0-15, 1: lanes 16-31
    const int sub  = lane & 15;
    const int khalf = half * 2;               // K pair this half-wave feeds

    const int aRow = tm * 16 + sub;           // A-matrix: lane -> row M
    const int bCol = tn * 16 + sub;           // B-matrix: lane -> col N

    // Column handling without control flow:
    const bool  colOK  = FULLN ? true : (bCol < N);
    const int   bColC  = FULLN ? bCol : (colOK ? bCol : (N - 1));
    const float bScale = colOK ? 1.0f : 0.0f;

    // Running pointers: no per-iteration 64-bit index math.
    const float* ap = A + (size_t)aRow * K + khalf;
    const float* bp = B + (size_t)khalf * N + bColC;
    const size_t bStep = (size_t)4 * N;

    v8f acc = {};

#if __has_builtin(__builtin_amdgcn_wmma_f32_16x16x4_f32)
#pragma unroll 8
    for (int k0 = 0; k0 < K; k0 += 4) {
        const float2 av = *(const float2*)ap;      // A: K pair, contiguous
        v2f a; a.x = av.x; a.y = av.y;
        v2f b;
        if (FULLN) {
            b.x = bp[0];
            b.y = bp[N];
        } else {
            b.x = bp[0] * bScale;
            b.y = bp[N] * bScale;
        }
        acc = __builtin_amdgcn_wmma_f32_16x16x4_f32(
            /*neg_a=*/false, a, /*neg_b=*/false, b,
            /*c_mod=*/(short)0, acc, /*reuse_a=*/false, /*reuse_b=*/false);
        ap += 4;
        bp += bStep;
    }
#else
    {   // Scalar fallback producing the same C VGPR layout (not expected)
        const int cRowBase = tm * 16 + half * 8;
        for (int r = 0; r < 8; ++r) {
            float s = 0.0f;
            if (colOK)
                for (int k = 0; k < K; ++k)
                    s += A[(size_t)(cRowBase + r) * K + k] * B[(size_t)k * N + bCol];
            acc[r] = s;
        }
    }
#endif

    // C/D 16x16 f32 layout: VGPR r -> row r (lanes 0-15) / row r+8 (lanes 16-31)
    const int cRowBase = tm * 16 + half * 8;
    if (colOK) {
        float* cp = C + (size_t)cRowBase * N + bCol;
#pragma unroll
        for (int r = 0; r < 8; ++r) {
            *cp = acc[r];
            cp += N;
        }
    }
}

// ---------------------------------------------------------------------------
// Zero-fill (float4 granularity; all buffer sizes here are multiples of 4)
// ---------------------------------------------------------------------------
__global__ __launch_bounds__(256) void zero_f32(float4* __restrict__ p, int n4)
{
    int i = blockIdx.x * blockDim.x + threadIdx.x;
    if (i < n4) p[i] = make_float4(0.f, 0.f, 0.f, 0.f);
}

// ---------------------------------------------------------------------------
// Edge scatter: agg[dst[e], :] += hw[src[e], :].  One thread per (edge,
// 4-feature chunk) -> coalesced float4 gather + 4 f32 atomics (L2-resident).
// ---------------------------------------------------------------------------
template <int F>
__global__ __launch_bounds__(256) void scatter_add(
    const float* __restrict__ hw, const long long* __restrict__ src,
    const long long* __restrict__ dst, float* __restrict__ agg)
{
    constexpr int F4 = F / 4;
    long long idx = (long long)blockIdx.x * blockDim.x + threadIdx.x;
    int e = (int)(idx / F4);
    int c = (int)(idx % F4);
    if (e >= EDGES) return;
    const int s = (int)src[e];
    const int d = (int)dst[e];
    const float4 v = *(const float4*)(hw + (size_t)s * F + c * 4);
    float* p = agg + (size_t)d * F + c * 4;
    atomicAdd(p + 0, v.x);
    atomicAdd(p + 1, v.y);
    atomicAdd(p + 2, v.z);
    atomicAdd(p + 3, v.w);
}

// ---------------------------------------------------------------------------
// out = (relu?) (agg + bias)
// ---------------------------------------------------------------------------
__global__ __launch_bounds__(256) void bias_act(
    const float* __restrict__ agg, const float* __restrict__ bias,
    float* __restrict__ out, int total, int F, int doRelu)
{
    int i = blockIdx.x * blockDim.x + threadIdx.x;
    if (i >= total) return;
    float v = agg[i] + bias[i % F];
    if (doRelu) v = fmaxf(v, 0.0f);
    out[i] = v;
}

// ---------------------------------------------------------------------------
extern "C" void kernel_launch(void* const* d_in, const int* in_sizes, int n_in,
                              void* d_out, int out_size, void* d_ws, size_t ws_size,
                              hipStream_t stream)
{
    (void)in_sizes; (void)n_in; (void)out_size; (void)ws_size;

    const float*     x   = (const float*)d_in[0];
    const long long* ei  = (const long long*)d_in[1];   // int64 [2, E]
    const float*     W0  = (const float*)d_in[2];
    const float*     b0  = (const float*)d_in[3];
    const float*     W1  = (const float*)d_in[4];
    const float*     b1  = (const float*)d_in[5];
    const float*     W2  = (const float*)d_in[6];
    const float*     b2  = (const float*)d_in[7];
    float*           out = (float*)d_out;

    const long long* src = ei;          // edge_index[0]
    const long long* dst = ei + EDGES;  // edge_index[1]

    // Workspace partition (all fp32): hW | agg | h   (3 x 25.6 MB, L2-resident)
    float* hW  = (float*)d_ws;
    float* agg = hW  + (size_t)NODES * 128;
    float* h   = agg + (size_t)NODES * 128;

    const dim3 blk(256);
    const int tilesM = NODES / 16;                 // 3125, exact

    // --------------------------- Layer 0 ---------------------------
    {
        const int tn = 128 / 16;                   // 8
        const int blocks = (tilesM * tn + 7) / 8;  // 3125
        gemm_f32_wmma<true><<<blocks, blk, 0, stream>>>(x, W0, hW, NODES, 128, 128, tn);

        const int n4 = NODES * 128 / 4;
        zero_f32<<<(n4 + 255) / 256, blk, 0, stream>>>((float4*)agg, n4);

        const long long st = (long long)EDGES * 32;
        scatter_add<128><<<(unsigned)((st + 255) / 256), blk, 0, stream>>>(hW, src, dst, agg);

        const int tot = NODES * 128;
        bias_act<<<(tot + 255) / 256, blk, 0, stream>>>(agg, b0, h, tot, 128, 1);
    }

    // --------------------------- Layer 1 ---------------------------
    {
        const int tn = 128 / 16;
        const int blocks = (tilesM * tn + 7) / 8;
        gemm_f32_wmma<true><<<blocks, blk, 0, stream>>>(h, W1, hW, NODES, 128, 128, tn);

        const int n4 = NODES * 128 / 4;
        zero_f32<<<(n4 + 255) / 256, blk, 0, stream>>>((float4*)agg, n4);

        const long long st = (long long)EDGES * 32;
        scatter_add<128><<<(unsigned)((st + 255) / 256), blk, 0, stream>>>(hW, src, dst, agg);

        const int tot = NODES * 128;
        bias_act<<<(tot + 255) / 256, blk, 0, stream>>>(agg, b1, h, tot, 128, 1);
    }

    // --------------------------- Layer 2 (N = 40) ---------------------------
    {
        const int tn = (40 + 15) / 16;             // 3 tiles, last one guarded
        const int blocks = (tilesM * tn + 7) / 8;  // 1172
        gemm_f32_wmma<false><<<blocks, blk, 0, stream>>>(h, W2, hW, NODES, 40, 128, tn);

        const int n4 = NODES * 40 / 4;
        zero_f32<<<(n4 + 255) / 256, blk, 0, stream>>>((float4*)agg, n4);

        const long long st = (long long)EDGES * 10;
        scatter_add<40><<<(unsigned)((st + 255) / 256), blk, 0, stream>>>(hW, src, dst, agg);

        const int tot = NODES * 40;
        bias_act<<<(tot + 255) / 256, blk, 0, stream>>>(agg, b2, out, tot, 40, 0);
    }
}